// TransformSinkhornRegModel_56358560858145
// MI455X (gfx1250) — compile-verified
//
#include <hip/hip_runtime.h>
#include <math.h>

typedef __attribute__((ext_vector_type(2))) float v2f;
typedef __attribute__((ext_vector_type(8))) float v8f;

#define ND 2048
#define NS 1024
#define DIN 768
#define DOUT 64
#define NSTEPS 20
#define BLUR2 0.0025f
#define SMWAVES 8

// ---------------------------------------------------------------------------
// Projection: x = d @ M (2048x64), y = s @ M (1024x64) via V_WMMA_F32_16X16X4_F32
// One wave per 16x16 output tile. Tiles 0..511 -> x, 512..767 -> y.
// ---------------------------------------------------------------------------
__global__ __launch_bounds__(256) void proj_kernel(
    const float* __restrict__ d, const float* __restrict__ s,
    const float* __restrict__ M, float* __restrict__ x, float* __restrict__ y) {
  const int lane = threadIdx.x & 31;
  const int wave = threadIdx.x >> 5;
  const int tile = blockIdx.x * 8 + wave;
  const float* src;
  float* dst;
  int mTile, nTile;
  if (tile < 512) { src = d; dst = x; mTile = tile >> 2; nTile = tile & 3; }
  else            { int tt = tile - 512; src = s; dst = y; mTile = tt >> 2; nTile = tt & 3; }
  const int l16 = lane & 15;
  const int half = lane >> 4;
  const float* Arow = src + (size_t)(mTile * 16 + l16) * DIN;
  const float* Bcol = M + (nTile * 16 + l16);  // column n, row stride DOUT

  v8f acc = {};
  for (int kk = 0; kk < DIN / 4; ++kk) {
    const int kb = kk * 4 + half * 2;
    v2f a = *(const v2f*)(Arow + kb);
    v2f b = { Bcol[(size_t)kb * DOUT], Bcol[(size_t)(kb + 1) * DOUT] };
    acc = __builtin_amdgcn_wmma_f32_16x16x4_f32(false, a, false, b,
                                                (short)0, acc, false, false);
  }
#pragma unroll
  for (int r = 0; r < 8; ++r) {
    const int row = mTile * 16 + r + 8 * half;
    dst[(size_t)row * DOUT + nTile * 16 + l16] = acc[r];
  }
}

// ---------------------------------------------------------------------------
// Squared row norms: sqx[i] = sum_k x[i][k]^2 ; same for y.
// ---------------------------------------------------------------------------
__global__ void rowstats_kernel(const float* __restrict__ x,
                                const float* __restrict__ y,
                                float* __restrict__ sqx, float* __restrict__ sqy) {
  const int i = blockIdx.x * blockDim.x + threadIdx.x;
  const float* p;
  float* o;
  if (i < ND) { p = x + (size_t)i * DOUT; o = sqx + i; }
  else if (i < ND + NS) { p = y + (size_t)(i - ND) * DOUT; o = sqy + (i - ND); }
  else return;
  float acc = 0.0f;
#pragma unroll 8
  for (int k = 0; k < DOUT; ++k) acc += p[k] * p[k];
  *o = acc;
}

// ---------------------------------------------------------------------------
// diam2 = sum_d (max_d - min_d)^2 over union of x,y rows. One block, 64 thr.
// Coalesced: threads = dims, loop over rows.
// ---------------------------------------------------------------------------
__global__ void diam_kernel(const float* __restrict__ x, const float* __restrict__ y,
                            float* __restrict__ diam2) {
  const int dIdx = threadIdx.x;  // 0..63
  float mn = INFINITY, mx = -INFINITY;
  for (int i = 0; i < ND; ++i) {
    float v = x[(size_t)i * DOUT + dIdx];
    mn = fminf(mn, v); mx = fmaxf(mx, v);
  }
  for (int i = 0; i < NS; ++i) {
    float v = y[(size_t)i * DOUT + dIdx];
    mn = fminf(mn, v); mx = fmaxf(mx, v);
  }
  __shared__ float red[64];
  red[dIdx] = (mx - mn) * (mx - mn);
  __syncthreads();
  for (int s2 = 32; s2 > 0; s2 >>= 1) {
    if (dIdx < s2) red[dIdx] += red[dIdx + s2];
    __syncthreads();
  }
  if (dIdx == 0) diam2[0] = red[0];
}

__global__ void zero_kernel(float* __restrict__ p, int n) {
  const int i = blockIdx.x * blockDim.x + threadIdx.x;
  if (i < n) p[i] = 0.0f;
}

// ---------------------------------------------------------------------------
// Fused softmin: out[i] = -eps * logsumexp_j((eps*logw + g[j] - C[i,j]) / eps)
// with C[i,j] = 0.5*sqx[i] + 0.5*sqy[j] - dot(X_i, Y_j).
// Gram tiles recomputed on the fly with f32 WMMA; online LSE accumulation.
// A fragments are loop-invariant and held in registers across the column loop.
// Grid: R/16 blocks; 8 waves/block split the column tiles; LDS combine.
// t >= 0 : eps = max(diam2 * 4^-t, blur^2) ; t < 0 : eps = blur^2.
// ---------------------------------------------------------------------------
__global__ __launch_bounds__(256) void softmin_kernel(
    const float* __restrict__ X, const float* __restrict__ sqx, int R,
    const float* __restrict__ Y, const float* __restrict__ sqy, int Ncol,
    const float* __restrict__ g, float* __restrict__ out,
    const float* __restrict__ diam2p, int t) {
  const int lane = threadIdx.x & 31;
  const int wave = threadIdx.x >> 5;
  const int l16 = lane & 15;
  const int half = lane >> 4;
  const int rowBase = blockIdx.x * 16;

  float eps;
  if (t >= 0) eps = fmaxf(diam2p[0] * exp2f(-2.0f * (float)t), BLUR2);
  else        eps = BLUR2;
  const float inv_eps = 1.0f / eps;
  const float logw = -__logf((float)Ncol);

  // Hoist loop-invariant A fragments: 16 x v2f = 32 VGPRs held live.
  const float* Arow = X + (size_t)(rowBase + l16) * DOUT;
  v2f aFrag[DOUT / 4];
#pragma unroll
  for (int kk = 0; kk < DOUT / 4; ++kk)
    aFrag[kk] = *(const v2f*)(Arow + kk * 4 + half * 2);

  float mAcc[8], sAcc[8];
#pragma unroll
  for (int r = 0; r < 8; ++r) { mAcc[r] = -INFINITY; sAcc[r] = 0.0f; }

  const int nTiles = Ncol >> 4;
  for (int ct = wave; ct < nTiles; ct += SMWAVES) {
    const int n = ct * 16 + l16;
    const float* Brow = Y + (size_t)n * DOUT;
    const float colterm = (g[n] - 0.5f * sqy[n]) * inv_eps;

    v8f acc = {};
#pragma unroll
    for (int kk = 0; kk < DOUT / 4; ++kk) {
      v2f b = *(const v2f*)(Brow + kk * 4 + half * 2);
      acc = __builtin_amdgcn_wmma_f32_16x16x4_f32(false, aFrag[kk], false, b,
                                                  (short)0, acc, false, false);
    }
#pragma unroll
    for (int r = 0; r < 8; ++r) {
      const float w = colterm + acc[r] * inv_eps;
      const float mNew = fmaxf(mAcc[r], w);
      sAcc[r] = sAcc[r] * __expf(mAcc[r] - mNew) + __expf(w - mNew);
      mAcc[r] = mNew;
    }
  }

  // Reduce LSE partials across the 16-lane half-groups (masks < 16 stay in-group).
#pragma unroll
  for (int k = 1; k < 16; k <<= 1) {
#pragma unroll
    for (int r = 0; r < 8; ++r) {
      const float mO = __shfl_xor(mAcc[r], k, 32);
      const float sO = __shfl_xor(sAcc[r], k, 32);
      const float mNew = fmaxf(mAcc[r], mO);
      sAcc[r] = sAcc[r] * __expf(mAcc[r] - mNew) + sO * __expf(mO - mNew);
      mAcc[r] = mNew;
    }
  }

  // Cross-wave combine in LDS. Lane 0 holds rows 0..7, lane 16 rows 8..15.
  __shared__ float ldsM[SMWAVES][16];
  __shared__ float ldsS[SMWAVES][16];
  if (l16 == 0) {
#pragma unroll
    for (int r = 0; r < 8; ++r) {
      ldsM[wave][half * 8 + r] = mAcc[r];
      ldsS[wave][half * 8 + r] = sAcc[r];
    }
  }
  __syncthreads();
  if (threadIdx.x < 16) {
    const int row = threadIdx.x;
    float m = ldsM[0][row], s = ldsS[0][row];
    for (int w2 = 1; w2 < SMWAVES; ++w2) {
      const float mO = ldsM[w2][row], sO = ldsS[w2][row];
      const float mNew = fmaxf(m, mO);
      s = s * __expf(m - mNew) + sO * __expf(mO - mNew);
      m = mNew;
    }
    const int i = rowBase + row;
    out[i] = -eps * (m + __logf(s)) - eps * logw + 0.5f * sqx[i];
  }
}

// pot[i] = mode ? 0.5*(pot[i]+tmp[i]) : tmp[i]   over all 4 potentials (6144).
__global__ void update_kernel(float* __restrict__ pot, const float* __restrict__ tmp,
                              int n, int mode) {
  const int i = blockIdx.x * blockDim.x + threadIdx.x;
  if (i >= n) return;
  pot[i] = mode ? 0.5f * (pot[i] + tmp[i]) : tmp[i];
}

// loss = mean(f_ab - f_aa) + mean(g_ab - g_bb); out = exp(-loss). One block.
__global__ void loss_kernel(const float* __restrict__ fab, const float* __restrict__ gab,
                            const float* __restrict__ faa, const float* __restrict__ gbb,
                            float* __restrict__ out) {
  float acc = 0.0f;
  for (int i = threadIdx.x; i < ND; i += 256) acc += (fab[i] - faa[i]) * (1.0f / (float)ND);
  for (int j = threadIdx.x; j < NS; j += 256) acc += (gab[j] - gbb[j]) * (1.0f / (float)NS);
  __shared__ float red[256];
  red[threadIdx.x] = acc;
  __syncthreads();
  for (int s2 = 128; s2 > 0; s2 >>= 1) {
    if (threadIdx.x < s2) red[threadIdx.x] += red[threadIdx.x + s2];
    __syncthreads();
  }
  if (threadIdx.x == 0) out[0] = expf(-red[0]);
}

// ---------------------------------------------------------------------------
extern "C" void kernel_launch(void* const* d_in, const int* in_sizes, int n_in,
                              void* d_out, int out_size, void* d_ws, size_t ws_size,
                              hipStream_t stream) {
  (void)in_sizes; (void)n_in; (void)out_size; (void)ws_size;
  const float* d = (const float*)d_in[0];  // [2048, 768]
  const float* s = (const float*)d_in[1];  // [1024, 768]
  const float* M = (const float*)d_in[2];  // [768, 64]
  float* out = (float*)d_out;

  float* ws   = (float*)d_ws;
  float* x    = ws;                         // 2048*64 = 131072
  float* y    = ws + 131072;                // 1024*64 =  65536
  float* sqx  = ws + 196608;                // 2048
  float* sqy  = ws + 198656;                // 1024
  float* pot  = ws + 199680;                // 6144: f_ab | g_ab | f_aa | g_bb
  float* tmp  = ws + 205824;                // 6144
  float* dia2 = ws + 211968;                // 1

  float* f_ab = pot + 0;
  float* g_ab = pot + 2048;
  float* f_aa = pot + 3072;
  float* g_bb = pot + 5120;

  proj_kernel<<<96, 256, 0, stream>>>(d, s, M, x, y);
  rowstats_kernel<<<12, 256, 0, stream>>>(x, y, sqx, sqy);
  diam_kernel<<<1, 64, 0, stream>>>(x, y, dia2);
  zero_kernel<<<24, 256, 0, stream>>>(pot, 6144);

  // --- init step at eps_list[0] (potentials are zero) ---
  softmin_kernel<<<128, 256, 0, stream>>>(x, sqx, ND, y, sqy, NS, g_ab, tmp + 0,    dia2, 0);
  softmin_kernel<<< 64, 256, 0, stream>>>(y, sqy, NS, x, sqx, ND, f_ab, tmp + 2048, dia2, 0);
  softmin_kernel<<<128, 256, 0, stream>>>(x, sqx, ND, x, sqx, ND, f_aa, tmp + 3072, dia2, 0);
  softmin_kernel<<< 64, 256, 0, stream>>>(y, sqy, NS, y, sqy, NS, g_bb, tmp + 5120, dia2, 0);
  update_kernel<<<24, 256, 0, stream>>>(pot, tmp, 6144, /*copy*/0);

  // --- eps-annealing scan: averaged updates, ft/gt read OLD potentials ---
  for (int t = 0; t < NSTEPS; ++t) {
    softmin_kernel<<<128, 256, 0, stream>>>(x, sqx, ND, y, sqy, NS, g_ab, tmp + 0,    dia2, t);
    softmin_kernel<<< 64, 256, 0, stream>>>(y, sqy, NS, x, sqx, ND, f_ab, tmp + 2048, dia2, t);
    softmin_kernel<<<128, 256, 0, stream>>>(x, sqx, ND, x, sqx, ND, f_aa, tmp + 3072, dia2, t);
    softmin_kernel<<< 64, 256, 0, stream>>>(y, sqy, NS, y, sqy, NS, g_bb, tmp + 5120, dia2, t);
    update_kernel<<<24, 256, 0, stream>>>(pot, tmp, 6144, /*avg*/1);
  }

  // --- final non-averaged step at eps = blur^2; g_ab uses the NEW f_ab ---
  softmin_kernel<<<128, 256, 0, stream>>>(x, sqx, ND, y, sqy, NS, g_ab, f_ab, dia2, -1);
  softmin_kernel<<< 64, 256, 0, stream>>>(y, sqy, NS, x, sqx, ND, f_ab, g_ab, dia2, -1);
  softmin_kernel<<<128, 256, 0, stream>>>(x, sqx, ND, x, sqx, ND, f_aa, tmp + 3072, dia2, -1);
  softmin_kernel<<< 64, 256, 0, stream>>>(y, sqy, NS, y, sqy, NS, g_bb, tmp + 5120, dia2, -1);

  loss_kernel<<<1, 256, 0, stream>>>(f_ab, g_ab, tmp + 3072, tmp + 5120, out);
}